// STN_Flow_relative_50895362458060
// MI455X (gfx1250) — compile-verified
//
#include <hip/hip_runtime.h>

// STN flow-relative bilinear grid sample, B=1, C=32, H=W=1024, fp32.
// Memory-bound gather: ~264 MiB min traffic -> ~11 us at 23.3 TB/s HBM.
// Strategy: one thread per output pixel; weights/indices computed once,
// channel loop does 4 gathers (GVS addressing: scalar plane base + 32-bit
// vector offset) + 4 FMAs. x (128 MiB) fits in the 192 MB L2; flow loads and
// output stores are non-temporal so they don't evict x.

#define CC 32
#define HH 1024
#define WW 1024

__global__ __launch_bounds__(256) void stn_flow_bilinear_kernel(
    const float* __restrict__ flow,   // [2, H, W]
    const float* __restrict__ x,      // [C, H, W]
    float* __restrict__ out)          // [C, H, W]
{
    const int HW = HH * WW;
    const int p  = blockIdx.x * blockDim.x + threadIdx.x;   // pixel id in [0, H*W)
    if (p >= HW) return;

    const int h = p >> 10;          // p / W
    const int w = p & (WW - 1);     // p % W

    // Stream flow ahead (gfx1250 global_prefetch_b8; speculative, OOB-safe).
    __builtin_prefetch(flow + p + 16384, 0, 0);
    __builtin_prefetch(flow + HW + p + 16384, 0, 0);

    // flow is read exactly once -> non-temporal load (keep L2 for x).
    const float f0 = __builtin_nontemporal_load(flow + p);        // x-displacement
    const float f1 = __builtin_nontemporal_load(flow + HW + p);   // y-displacement

    // Replicate reference math order exactly (align_corners=False).
    const float gx = (f0 + (float)w) / (float)(WW - 1) * 2.0f - 1.0f;
    const float gy = (f1 + (float)h) / (float)(HH - 1) * 2.0f - 1.0f;
    const float ix = ((gx + 1.0f) * (float)WW - 1.0f) * 0.5f;
    const float iy = ((gy + 1.0f) * (float)HH - 1.0f) * 0.5f;

    const float x0f = floorf(ix);
    const float y0f = floorf(iy);
    const float wx1 = ix - x0f;
    const float wy1 = iy - y0f;
    const float wx0 = 1.0f - wx1;
    const float wy0 = 1.0f - wy1;

    const int x0 = (int)x0f;
    const int y0 = (int)y0f;
    const int x1 = x0 + 1;
    const int y1 = y0 + 1;

    // Zero-padding: fold per-corner validity masks into the weights.
    const float mwx0 = (x0 >= 0 && x0 <= WW - 1) ? wx0 : 0.0f;
    const float mwx1 = (x1 >= 0 && x1 <= WW - 1) ? wx1 : 0.0f;
    const float mwy0 = (y0 >= 0 && y0 <= HH - 1) ? wy0 : 0.0f;
    const float mwy1 = (y1 >= 0 && y1 <= HH - 1) ? wy1 : 0.0f;

    const float w00 = mwx0 * mwy0;
    const float w10 = mwx1 * mwy0;
    const float w01 = mwx0 * mwy1;
    const float w11 = mwx1 * mwy1;

    // Clamped indices for the actual loads (masked corners contribute 0 anyway).
    const int x0c = min(max(x0, 0), WW - 1);
    const int x1c = min(max(x1, 0), WW - 1);
    const int y0c = min(max(y0, 0), HH - 1);
    const int y1c = min(max(y1, 0), HH - 1);

    // 32-bit vector offsets within a channel plane (max 4 MiB -> fits i32,
    // enables saddr(scalar plane base) + voffset global loads).
    const int o00 = y0c * WW + x0c;
    const int o10 = y0c * WW + x1c;
    const int o01 = y1c * WW + x0c;
    const int o11 = y1c * WW + x1c;

    const float* plane = x;
    float*       op    = out + p;

#pragma unroll 4
    for (int c = 0; c < CC; ++c) {
        const float v00 = plane[o00];
        const float v10 = plane[o10];
        const float v01 = plane[o01];
        const float v11 = plane[o11];
        const float r = v00 * w00 + v10 * w10 + v01 * w01 + v11 * w11;
        // Output written once, never re-read -> non-temporal store.
        __builtin_nontemporal_store(r, op);
        plane += HW;   // scalar stride: next channel plane
        op    += HW;
    }
}

extern "C" void kernel_launch(void* const* d_in, const int* in_sizes, int n_in,
                              void* d_out, int out_size, void* d_ws, size_t ws_size,
                              hipStream_t stream) {
    const float* flow = (const float*)d_in[0];   // [1,2,1024,1024]
    const float* x    = (const float*)d_in[1];   // [1,32,1024,1024]
    float*       out  = (float*)d_out;           // [1,32,1024,1024]

    const int HW = HH * WW;
    const int threads = 256;                     // 8 wave32 waves per block
    const int blocks  = (HW + threads - 1) / threads;
    stn_flow_bilinear_kernel<<<blocks, threads, 0, stream>>>(flow, x, out);
}